// DHSNN_86474871538214
// MI455X (gfx1250) — compile-verified
//
#include <hip/hip_runtime.h>
#include <math.h>

#define T_STEPS 250
#define NBATCH  256
#define INDIM   1024
#define HDIM    1024
#define OUTDIM  35
#define NBRANCH 4
#define INB     256
#define TCHUNK  25
#define NCHUNK  (T_STEPS / TCHUNK)
#define BPITCH  80            // LDS row pitch in dwords: 64 data + 16 pad (bank-conflict-free)

typedef __attribute__((ext_vector_type(2))) float v2f;
typedef __attribute__((ext_vector_type(4))) float v4f;
typedef __attribute__((ext_vector_type(8))) float v8f;
typedef __attribute__((ext_vector_type(4))) unsigned int u32x4;
typedef __attribute__((ext_vector_type(8))) int i32x8;
typedef __attribute__((ext_vector_type(4))) int i32x4;

// ---------------------------------------------------------------------------
// Phase 1: batched branch GEMM for a time chunk.
//   Y[m, b, h] = sum_k x[row_base + m, b*256 + k] * Wb[b, k, h]
// Block tile 128(M) x 64(H); the whole K=256 x 64 B-slab is staged into LDS
// once per block via the Tensor Data Mover (padded rows -> conflict-free
// ds reads), then each of 8 waves computes a 16x64 tile with
// V_WMMA_F32_16X16X4_F32. grid = (Mc/128, HDIM/64, NBRANCH), block = 256.
// ---------------------------------------------------------------------------
__global__ __launch_bounds__(256) void dhsnn_branch_gemm(
    const float* __restrict__ x, const float* __restrict__ Wb,
    float* __restrict__ Y, int row_base)
{
    extern __shared__ float Bs[];        // [256][BPITCH] dwords = 80 KB dynamic LDS

    const int tid  = threadIdx.x;
    const int lane = tid & 31;
    const int wave = tid >> 5;
    const int half = lane >> 4;          // 0: lanes 0-15 (K0,K1) | 1: lanes 16-31 (K2,K3)
    const int l16  = lane & 15;
    const int b     = blockIdx.z;
    const int hbase = blockIdx.y * 64;
    const int mrow  = blockIdx.x * 128 + wave * 16;   // row within chunk

    const float* Wg = Wb + (size_t)b * INB * HDIM + hbase;   // tile start in Wb

#if __has_builtin(__builtin_amdgcn_tensor_load_to_lds)
    // TDM: DMA the 256x64 f32 tile (row stride 1024 elements) into LDS with
    // 16-dword padding after every 64-dword row (pitch = 80 dwords).
    if (tid < 32) {                       // wave 0 issues once (TDM ignores EXEC)
        unsigned long long ga = (unsigned long long)(const void*)Wg;
        u32x4 g0;
        g0[0] = 1u;                                   // count = 1 valid descriptor
        g0[1] = 0u;                                   // lds_addr = 0 (Bs is only LDS object)
        g0[2] = (unsigned)(ga & 0xFFFFFFFFu);         // global_addr[31:0]
        g0[3] = (unsigned)((ga >> 32) & 0x1FFFFFFu)   // global_addr[56:32]
              | (2u << 30);                           // type = 2 ("image")
        i32x8 g1;
        g1[0] = (int)((2u << 16)                      // data_size = 4 bytes
              | (1u << 20)                            // pad_enable
              | (5u << 22)                            // pad_interval: 64 dwords
              | (15u << 25));                         // pad_amount: 16 dwords
        g1[1] = (int)(1024u << 16);                   // tensor_dim0 = 1024 (row length)
        g1[2] = (int)(256u << 16);                    // tensor_dim1 = 256 (rows)
        g1[3] = (int)(64u << 16);                     // tile_dim0 = 64
        g1[4] = 256;                                  // tile_dim1 = 256
        g1[5] = 1024;                                 // tensor_dim0_stride = 1024 elements
        g1[6] = 0;
        g1[7] = 0;
        i32x4 z4 = {0, 0, 0, 0};
        i32x8 z8 = {0, 0, 0, 0, 0, 0, 0, 0};
        __builtin_amdgcn_tensor_load_to_lds(g0, g1, z4, z4, z8, 0);
        __builtin_amdgcn_s_wait_tensorcnt(0);
    }
#else
    // Fallback: cooperative load producing the identical padded LDS layout.
    for (int idx = tid; idx < INB * 64; idx += 256) {
        int k = idx >> 6, col = idx & 63;
        Bs[k * BPITCH + col] = Wg[(size_t)k * HDIM + col];
    }
#endif
    __syncthreads();

    // A fragment base: lane l16 holds row (mrow+l16); half selects K pair.
    const float* arow = x + (size_t)(row_base + mrow + l16) * INDIM + b * INB + half * 2;
    // B fragments from LDS: VGPR0 = row (k0 + 2*half), VGPR1 = next row.
    const float* bs = Bs + (half * 2) * BPITCH + l16;

    v8f acc0 = {}, acc1 = {}, acc2 = {}, acc3 = {};

    for (int k0 = 0; k0 < INB; k0 += 4) {
        v2f a = *(const v2f*)(arow + k0);            // contiguous K pair, 8B aligned
        const float* bp = bs + k0 * BPITCH;
        v2f b0, b1, b2, b3;
        b0.x = bp[0];   b0.y = bp[BPITCH];
        b1.x = bp[16];  b1.y = bp[16 + BPITCH];
        b2.x = bp[32];  b2.y = bp[32 + BPITCH];
        b3.x = bp[48];  b3.y = bp[48 + BPITCH];
        acc0 = __builtin_amdgcn_wmma_f32_16x16x4_f32(false, a, false, b0, (short)0, acc0, false, false);
        acc1 = __builtin_amdgcn_wmma_f32_16x16x4_f32(false, a, false, b1, (short)0, acc1, false, false);
        acc2 = __builtin_amdgcn_wmma_f32_16x16x4_f32(false, a, false, b2, (short)0, acc2, false, false);
        acc3 = __builtin_amdgcn_wmma_f32_16x16x4_f32(false, a, false, b3, (short)0, acc3, false, false);
    }

    // C/D layout: VGPR r -> row (r + 8*half), col = l16 (+16j per subtile).
    float* yp = Y + (size_t)(mrow + half * 8) * (NBRANCH * HDIM) + b * HDIM + hbase + l16;
#pragma unroll
    for (int r = 0; r < 8; ++r) {
        size_t ro = (size_t)r * (NBRANCH * HDIM);
        yp[ro + 0 ] = acc0[r];
        yp[ro + 16] = acc1[r];
        yp[ro + 32] = acc2[r];
        yp[ro + 48] = acc3[r];
    }
}

// ---------------------------------------------------------------------------
// Phase 2: persistent per-batch-element recurrence over one time chunk.
// One block per n; each thread owns 4 hidden lanes and all 4 branches of
// dendritic state in registers. Spike-gated GEMV s1 @ W2 with wave-uniform h.
// ---------------------------------------------------------------------------
__global__ __launch_bounds__(256) void dhsnn_recur(
    const float* __restrict__ Y,
    const float* __restrict__ bb, const float* __restrict__ taus,
    const float* __restrict__ W2, const float* __restrict__ b2,
    const float* __restrict__ tau1p, const float* __restrict__ tau2p,
    float* __restrict__ stg, float* __restrict__ v1g,
    float* __restrict__ v2g, float* __restrict__ accg,
    float* __restrict__ out, int t0, int last)
{
    __shared__ float s1s[HDIM];
    __shared__ float h2s[OUTDIM];
    __shared__ float red[1];

    const int n    = blockIdx.x;
    const int tid  = threadIdx.x;
    const int h0   = tid * 4;
    const int wave = tid >> 5;
    const int lane = tid & 31;

    const float inv_t1 = 1.0f / tau1p[0];
    const float inv_t2 = 1.0f / tau2p[0];

    float alp[NBRANCH][4], bbr[NBRANCH][4];
#pragma unroll
    for (int b = 0; b < NBRANCH; ++b)
#pragma unroll
        for (int j = 0; j < 4; ++j) {
            float tv = taus[b * HDIM + h0 + j];
            alp[b][j] = 1.0f / (1.0f + expf(-tv));   // sigmoid gate
            bbr[b][j] = bb[b * HDIM + h0 + j];
        }

    float st[NBRANCH][4];
    float v1[4];
    float v2 = 0.0f, accv = 0.0f;
    const float b2r = (tid < OUTDIM) ? b2[tid] : 0.0f;

    if (t0 == 0) {
#pragma unroll
        for (int b = 0; b < NBRANCH; ++b)
#pragma unroll
            for (int j = 0; j < 4; ++j) st[b][j] = 0.0f;
#pragma unroll
        for (int j = 0; j < 4; ++j) v1[j] = 0.0f;
    } else {
#pragma unroll
        for (int b = 0; b < NBRANCH; ++b)
#pragma unroll
            for (int j = 0; j < 4; ++j)
                st[b][j] = stg[((size_t)n * NBRANCH + b) * HDIM + h0 + j];
#pragma unroll
        for (int j = 0; j < 4; ++j) v1[j] = v1g[(size_t)n * HDIM + h0 + j];
        if (tid < OUTDIM) { v2 = v2g[n * OUTDIM + tid]; accv = accg[n * OUTDIM + tid]; }
    }

    for (int tt = 0; tt < TCHUNK; ++tt) {
        if (tid < OUTDIM) h2s[tid] = 0.0f;

        const float* Yt = Y + ((size_t)tt * NBATCH + n) * (NBRANCH * HDIM);
        float comb[4] = {0.0f, 0.0f, 0.0f, 0.0f};
#pragma unroll
        for (int b = 0; b < NBRANCH; ++b) {
            v4f yv = *(const v4f*)(Yt + b * HDIM + h0);
#pragma unroll
            for (int j = 0; j < 4; ++j) {
                float bi = yv[j] + bbr[b][j];
                st[b][j] = alp[b][j] * st[b][j] + (1.0f - alp[b][j]) * bi;
                comb[j] += st[b][j];
            }
        }
#pragma unroll
        for (int j = 0; j < 4; ++j) {
            v1[j] += (comb[j] - v1[j]) * inv_t1;
            float s = (v1[j] >= 1.0f) ? 1.0f : 0.0f;   // heaviside(v1 - 1)
            v1[j] *= (1.0f - s);                        // hard reset
            s1s[h0 + j] = s;
        }
        __syncthreads();

        // Spike-gated GEMV: wave `wave` covers h in [wave*128, wave*128+128).
        // All 32 lanes share each h -> s1 gate is wave-uniform (no divergence).
        float a0 = 0.0f, a1 = 0.0f;
        const int hb = wave << 7;
        for (int hh = 0; hh < 128; hh += 4) {
            v4f s = *(const v4f*)&s1s[hb + hh];        // ds_load_b128 broadcast
#pragma unroll
            for (int j = 0; j < 4; ++j) {
                if (s[j] != 0.0f) {
                    int h = hb + hh + j;
                    a0 += W2[h * OUTDIM + lane];                     // outputs 0..31
                    if (lane < 3) a1 += W2[h * OUTDIM + 32 + lane];  // outputs 32..34
                }
            }
        }
        atomicAdd(&h2s[lane], a0);                      // ds_add_f32
        if (lane < 3) atomicAdd(&h2s[32 + lane], a1);
        __syncthreads();

        if (tid < OUTDIM) {
            float h2 = h2s[tid] + b2r;
            v2 += (h2 - v2) * inv_t2;
            float s2 = (v2 >= 1.0f) ? 1.0f : 0.0f;
            v2 *= (1.0f - s2);
            accv += s2;
        }
        __syncthreads();
    }

    // Persist carry state for next chunk.
#pragma unroll
    for (int b = 0; b < NBRANCH; ++b)
#pragma unroll
        for (int j = 0; j < 4; ++j)
            stg[((size_t)n * NBRANCH + b) * HDIM + h0 + j] = st[b][j];
#pragma unroll
    for (int j = 0; j < 4; ++j) v1g[(size_t)n * HDIM + h0 + j] = v1[j];
    if (tid < OUTDIM) { v2g[n * OUTDIM + tid] = v2; accg[n * OUTDIM + tid] = accv; }

    if (last) {
        if (tid < OUTDIM) h2s[tid] = accv;
        __syncthreads();
        if (tid == 0) {
            float m = -3.402823466e38f;
            for (int o = 0; o < OUTDIM; ++o) m = fmaxf(m, h2s[o]);
            float s = 0.0f;
            for (int o = 0; o < OUTDIM; ++o) s += expf(h2s[o] - m);
            red[0] = m + logf(s);
        }
        __syncthreads();
        if (tid < OUTDIM) out[n * OUTDIM + tid] = accv - red[0];
    }
}

// ---------------------------------------------------------------------------
extern "C" void kernel_launch(void* const* d_in, const int* in_sizes, int n_in,
                              void* d_out, int out_size, void* d_ws, size_t ws_size,
                              hipStream_t stream) {
    (void)in_sizes; (void)n_in; (void)out_size; (void)ws_size;
    const float* x    = (const float*)d_in[0];
    const float* Wb   = (const float*)d_in[1];
    const float* bb   = (const float*)d_in[2];
    const float* taus = (const float*)d_in[3];
    const float* W2   = (const float*)d_in[4];
    const float* b2   = (const float*)d_in[5];
    const float* tau1 = (const float*)d_in[6];
    const float* tau2 = (const float*)d_in[7];
    float* out = (float*)d_out;

    // Workspace: [Y chunk: TCHUNK*N*B*H f32 (~105 MB, L2-resident)] [state] [v1] [v2] [acc]
    char* ws = (char*)d_ws;
    size_t szY = (size_t)TCHUNK * NBATCH * NBRANCH * HDIM * sizeof(float);
    float* Y    = (float*)ws;
    float* stg  = (float*)(ws + szY);
    float* v1g  = stg + (size_t)NBATCH * NBRANCH * HDIM;
    float* v2g  = v1g + (size_t)NBATCH * HDIM;
    float* accg = v2g + (size_t)NBATCH * OUTDIM;

    const size_t ldsB = (size_t)INB * BPITCH * sizeof(float);   // 80 KB
    dim3 ggrid(TCHUNK * NBATCH / 128, HDIM / 64, NBRANCH);      // (50, 16, 4)
    for (int c = 0; c < NCHUNK; ++c) {
        int t0 = c * TCHUNK;
        dhsnn_branch_gemm<<<ggrid, dim3(256), ldsB, stream>>>(x, Wb, Y, t0 * NBATCH);
        dhsnn_recur<<<dim3(NBATCH), dim3(256), 0, stream>>>(
            Y, bb, taus, W2, b2, tau1, tau2,
            stg, v1g, v2g, accg, out, t0, (c == NCHUNK - 1) ? 1 : 0);
    }
}